// DirectedDualSAGE_30339648979103
// MI455X (gfx1250) — compile-verified
//
#include <hip/hip_runtime.h>
#include <hip/hip_bf16.h>

// ---------------------------------------------------------------------------
// DirectedDualSAGE on MI455X (gfx1250, wave32).
// GEMMs: V_WMMA_F32_16X16X4_F32 with A-operands staged in LDS (conflict-free
// padded pitch), B-operands from global (L0/L2-resident weights, no intra-
// block redundancy). Aggregation: wave-per-edge scatter, f32 atomics in L2.
// ---------------------------------------------------------------------------

typedef float v2f __attribute__((ext_vector_type(2)));
typedef float v8f __attribute__((ext_vector_type(8)));

#define NNODES 50000
#define FEAT   128
#define HALF   64
#define MTILES (NNODES / 16)   // 3125 (N divisible by 16 -> EXEC all-1s at WMMA)
#define SPAD   132             // LDS row pitch for 128-wide tiles: bank = (4m+k)%64
#define CPAD   260             // LDS row pitch for 256-wide tiles: 260%64 == 4

// -------------------- utility: zero fill --------------------
__global__ void fill_zero_kernel(float* __restrict__ p, long long count) {
  long long i = (long long)blockIdx.x * blockDim.x + threadIdx.x;
  if (i < count) p[i] = 0.0f;
}

// -------------------- degree (scatter count) --------------------
__global__ void degree_kernel(const int* __restrict__ dst, float* __restrict__ deg, int E) {
  int e = blockIdx.x * blockDim.x + threadIdx.x;
  if (e < E) atomicAdd(&deg[dst[e]], 1.0f);
}

// -------------------- scatter-add of features --------------------
// One wave32 per edge: src/dst uniform per wave (scalar broadcast loads),
// 32 lanes x float4 cover the 128-feature row -> coalesced 512B gather,
// 128 contiguous f32 atomics into msum[dst] (L2-resident).
__global__ void scatter_kernel(const float* __restrict__ X,
                               const int* __restrict__ src,
                               const int* __restrict__ dst,
                               float* __restrict__ msum, int E) {
  long long t = (long long)blockIdx.x * blockDim.x + threadIdx.x;
  int e = (int)(t >> 5);
  if (e >= E) return;
  int f = ((int)t & 31) * 4;
  int s = src[e];
  int d = dst[e];
  const float4 v = *(const float4*)(X + (size_t)s * FEAT + f);
  float* mp = msum + (size_t)d * FEAT + f;
  atomicAdd(mp + 0, v.x);
  atomicAdd(mp + 1, v.y);
  atomicAdd(mp + 2, v.z);
  atomicAdd(mp + 3, v.w);
}

// -------------------- SAGE conv GEMM --------------------
// h[m][n] = ReLU( (msum[m]/max(deg[m],1)) @ Wl + H[m] @ Wr + bl[n] ), n in [0,64)
// grid.x = MTILES (16 rows), block = 128 = 4 waves; wave w -> 16-col tile w.
// A fragments come from LDS (staged once, mean pre-scaled); B from global.
__global__ __launch_bounds__(128) void sage_gemm_kernel(
    const float* __restrict__ msum, const float* __restrict__ deg,
    const float* __restrict__ H, const float* __restrict__ Wl,
    const float* __restrict__ Wr, const float* __restrict__ bl,
    float* __restrict__ out) {
  __shared__ float lds_mean[16 * SPAD];
  __shared__ float lds_h[16 * SPAD];

  const int tile_m = blockIdx.x;
  const int tid    = threadIdx.x;

  // ---- stage A tiles: 16 rows x 128 cols x 2 matrices (512 float4 each) ----
#pragma unroll
  for (int i = 0; i < 4; ++i) {
    const int idx = tid + i * 128;        // 0..511
    const int row = idx >> 5;             // 0..15
    const int c4  = (idx & 31) * 4;       // 0,4,...,124
    const int gm  = tile_m * 16 + row;
    const float rs = 1.0f / fmaxf(deg[gm], 1.0f);
    const float4 mv = *(const float4*)(msum + (size_t)gm * FEAT + c4);
    const float4 hv = *(const float4*)(H    + (size_t)gm * FEAT + c4);
    float4 sv = { mv.x * rs, mv.y * rs, mv.z * rs, mv.w * rs };
    *(float4*)&lds_mean[row * SPAD + c4] = sv;
    *(float4*)&lds_h[row * SPAD + c4]    = hv;
  }
  __syncthreads();

  const int wave = tid >> 5;
  const int lane = tid & 31;
  const int mi   = lane & 15;     // A row within tile
  const int h    = lane >> 4;     // K-pair select (A/B), row-pair select (C/D)
  const int n    = wave * 16 + mi;

  v8f c = {};
#pragma unroll 8
  for (int kc = 0; kc < FEAT; kc += 4) {
    const int k = kc + 2 * h;
    v2f a1 = *(const v2f*)&lds_mean[mi * SPAD + k];
    v2f b1 = { Wl[(size_t)k * HALF + n], Wl[(size_t)(k + 1) * HALF + n] };
    c = __builtin_amdgcn_wmma_f32_16x16x4_f32(false, a1, false, b1, (short)0, c, false, false);
    v2f a2 = *(const v2f*)&lds_h[mi * SPAD + k];
    v2f b2 = { Wr[(size_t)k * HALF + n], Wr[(size_t)(k + 1) * HALF + n] };
    c = __builtin_amdgcn_wmma_f32_16x16x4_f32(false, a2, false, b2, (short)0, c, false, false);
  }

  const float bias  = bl[n];
  const int   mbase = tile_m * 16 + 8 * h;   // C/D: VGPR r -> row r + 8*h
#pragma unroll
  for (int r = 0; r < 8; ++r) {
    float v = c[r] + bias;
    out[(size_t)(mbase + r) * HALF + n] = fmaxf(v, 0.0f);
  }
}

// -------------------- combine GEMM --------------------
// out[m][n] = ReLU( [X | Hin | Hout][m][0:256] @ W[256][128] + b[n] )
// grid.x = MTILES, block = 256 = 8 waves; wave w -> 16-col tile w.
// Concat A tile staged in LDS once (16 x 256, pitch 260); B from global.
__global__ __launch_bounds__(256) void comb_gemm_kernel(
    const float* __restrict__ X, const float* __restrict__ Hin,
    const float* __restrict__ Hout, const float* __restrict__ W,
    const float* __restrict__ b, float* __restrict__ out) {
  __shared__ float lds_a[16 * CPAD];

  const int tile_m = blockIdx.x;
  const int tid    = threadIdx.x;

  // ---- stage concat A tile: 16 rows x 256 cols (1024 float4) ----
#pragma unroll
  for (int i = 0; i < 4; ++i) {
    const int idx = tid + i * 256;        // 0..1023
    const int row = idx >> 6;             // 0..15  (64 float4 per row)
    const int c4  = (idx & 63) * 4;       // 0,4,...,252
    const int gm  = tile_m * 16 + row;
    float4 v;
    if (c4 < 128)      v = *(const float4*)(X    + (size_t)gm * FEAT + c4);
    else if (c4 < 192) v = *(const float4*)(Hin  + (size_t)gm * HALF + (c4 - 128));
    else               v = *(const float4*)(Hout + (size_t)gm * HALF + (c4 - 192));
    *(float4*)&lds_a[row * CPAD + c4] = v;
  }
  __syncthreads();

  const int wave = tid >> 5;
  const int lane = tid & 31;
  const int mi   = lane & 15;
  const int h    = lane >> 4;
  const int n    = wave * 16 + mi;

  v8f c = {};
#pragma unroll 8
  for (int kc = 0; kc < 2 * FEAT; kc += 4) {
    const int k = kc + 2 * h;
    v2f a  = *(const v2f*)&lds_a[mi * CPAD + k];
    v2f bb = { W[(size_t)k * FEAT + n], W[(size_t)(k + 1) * FEAT + n] };
    c = __builtin_amdgcn_wmma_f32_16x16x4_f32(false, a, false, bb, (short)0, c, false, false);
  }

  const float bias  = b[n];
  const int   mbase = tile_m * 16 + 8 * h;
#pragma unroll
  for (int r = 0; r < 8; ++r) {
    float v = c[r] + bias;
    out[(size_t)(mbase + r) * FEAT + n] = fmaxf(v, 0.0f);
  }
}

// -------------------- final projection: out[i] = H[i,:] . W + b --------------------
__global__ void final_kernel(const float* __restrict__ H, const float* __restrict__ W,
                             const float* __restrict__ b, float* __restrict__ out, int n) {
  int i = blockIdx.x * blockDim.x + threadIdx.x;
  if (i >= n) return;
  const float4* hr = (const float4*)(H + (size_t)i * FEAT);
  const float4* wr = (const float4*)W;
  float acc = 0.0f;
#pragma unroll 8
  for (int k = 0; k < FEAT / 4; ++k) {
    float4 hv = hr[k];
    float4 wv = wr[k];
    acc += hv.x * wv.x + hv.y * wv.y + hv.z * wv.z + hv.w * wv.w;
  }
  out[i] = acc + b[0];
}

// ---------------------------------------------------------------------------
extern "C" void kernel_launch(void* const* d_in, const int* in_sizes, int n_in,
                              void* d_out, int out_size, void* d_ws, size_t ws_size,
                              hipStream_t stream) {
  (void)n_in; (void)out_size; (void)ws_size;

  const float* x       = (const float*)d_in[0];
  const int*   ein     = (const int*)d_in[1];   // [2,E]: row0=src, row1=dst
  const int*   eout    = (const int*)d_in[2];
  const float* in_Wl0  = (const float*)d_in[3];
  const float* in_bl0  = (const float*)d_in[4];
  const float* in_Wr0  = (const float*)d_in[5];
  const float* out_Wl0 = (const float*)d_in[6];
  const float* out_bl0 = (const float*)d_in[7];
  const float* out_Wr0 = (const float*)d_in[8];
  const float* comb_W0 = (const float*)d_in[9];
  const float* comb_b0 = (const float*)d_in[10];
  const float* in_Wl1  = (const float*)d_in[11];
  const float* in_bl1  = (const float*)d_in[12];
  const float* in_Wr1  = (const float*)d_in[13];
  const float* out_Wl1 = (const float*)d_in[14];
  const float* out_bl1 = (const float*)d_in[15];
  const float* out_Wr1 = (const float*)d_in[16];
  const float* comb_W1 = (const float*)d_in[17];
  const float* comb_b1 = (const float*)d_in[18];
  const float* final_W = (const float*)d_in[19];
  const float* final_b = (const float*)d_in[20];
  float* out = (float*)d_out;

  const int N = NNODES;
  const int E = in_sizes[1] / 2;  // 625000

  // ---- workspace layout (floats) ----
  const size_t N128 = (size_t)N * FEAT;
  const size_t N64  = (size_t)N * HALF;
  const size_t PADN = 65536;  // padded slot for N=50000 degree arrays
  float* ws     = (float*)d_ws;
  float* degIn  = ws;
  float* degOut = ws + PADN;
  float* msum   = ws + 2 * PADN;
  float* hin    = msum + N128;
  float* hout   = hin + N64;
  float* HA     = hout + N64;
  float* HB     = HA + N128;

  const int ZT = 256;
  auto nblk = [](long long c, int t) { return (unsigned)((c + t - 1) / t); };

  // ---- degrees (edge-list dependent only; shared by both layers) ----
  fill_zero_kernel<<<nblk(N, ZT), ZT, 0, stream>>>(degIn, N);
  fill_zero_kernel<<<nblk(N, ZT), ZT, 0, stream>>>(degOut, N);
  degree_kernel<<<nblk(E, ZT), ZT, 0, stream>>>(ein + E, degIn, E);
  degree_kernel<<<nblk(E, ZT), ZT, 0, stream>>>(eout + E, degOut, E);

  auto conv = [&](const float* H, const int* edges, const float* deg,
                  const float* Wl, const float* bl, const float* Wr, float* hdst) {
    fill_zero_kernel<<<nblk((long long)N128, ZT), ZT, 0, stream>>>(msum, (long long)N128);
    scatter_kernel<<<nblk((long long)E * 32, ZT), ZT, 0, stream>>>(H, edges, edges + E, msum, E);
    sage_gemm_kernel<<<MTILES, 128, 0, stream>>>(msum, deg, H, Wl, Wr, bl, hdst);
  };

  // ---- layer 0 (input = x) ----
  conv(x, ein,  degIn,  in_Wl0,  in_bl0,  in_Wr0,  hin);
  conv(x, eout, degOut, out_Wl0, out_bl0, out_Wr0, hout);
  comb_gemm_kernel<<<MTILES, 256, 0, stream>>>(x, hin, hout, comb_W0, comb_b0, HA);

  // ---- layer 1 (input = HA) ----
  conv(HA, ein,  degIn,  in_Wl1,  in_bl1,  in_Wr1,  hin);
  conv(HA, eout, degOut, out_Wl1, out_bl1, out_Wr1, hout);
  comb_gemm_kernel<<<MTILES, 256, 0, stream>>>(HA, hin, hout, comb_W1, comb_b1, HB);

  // ---- final projection ----
  final_kernel<<<nblk(N, ZT), ZT, 0, stream>>>(HB, final_W, final_b, out, N);
}